// SKTCell_78194174591489
// MI455X (gfx1250) — compile-verified
//
#include <hip/hip_runtime.h>
#include <hip/hip_bf16.h>

// Shapes
#define BATCH 32
#define N_EX  1024
#define N_KC  128
#define D_E   64
#define D_H   64
#define D3    192   // 3*D_H

typedef __attribute__((ext_vector_type(2))) float v2f;
typedef __attribute__((ext_vector_type(8))) float v8f;

__device__ __forceinline__ float sigmoidf_(float x) {
    return 1.0f / (1.0f + __expf(-x));
}

// ---------------------------------------------------------------------------
// CDNA5 async global->LDS copy (ASYNCcnt-tracked, bypasses VGPRs).
// lds_addr: LDS byte offset (low 32 bits of flat shared pointer per ISA
// aperture rule); gaddr: 64-bit global address in a VGPR pair.
// ---------------------------------------------------------------------------
__device__ __forceinline__ void async_ld_b128(unsigned lds_addr,
                                              unsigned long long gaddr) {
    asm volatile("global_load_async_to_lds_b128 %0, %1, off"
                 :: "v"(lds_addr), "v"(gaddr)
                 : "memory");
}
__device__ __forceinline__ void wait_async0() {
    asm volatile("s_wait_asynccnt 0x0" ::: "memory");
}

// ---------------------------------------------------------------------------
// Kernel 1: small precompute (VALU): kc_graph = sigmoid(kc_gamma),
//           x = x_id @ W_ex.T, kci = ex @ ex_graph, epart = e @ W_e.T + b
// Flat job dispatch: [0,16384) kcg | [16384,18432) x | [18432,22528) kci |
//                    [22528,30720) epart
// ---------------------------------------------------------------------------
__global__ void skt_pre(const float* __restrict__ kc_gamma,
                        const float* __restrict__ ex,
                        const float* __restrict__ su,
                        const float* __restrict__ W_ex,
                        const float* __restrict__ ex_graph,
                        const float* __restrict__ W_kc,
                        const float* __restrict__ fpart_w,
                        const float* __restrict__ fpart_b,
                        float* __restrict__ kcg,
                        float* __restrict__ xw,
                        float* __restrict__ kciw,
                        float* __restrict__ epartw) {
    int idx = blockIdx.x * blockDim.x + threadIdx.x;
    if (idx < 16384) {
        kcg[idx] = sigmoidf_(kc_gamma[idx]);
    } else if (idx < 18432) {
        int t = idx - 16384;            // x[b,e], b in [0,32), e in [0,64)
        int b = t >> 6, e = t & 63;
        const float* exb = ex + b * N_EX;
        const float* w0  = W_ex + e * (2 * N_EX);
        const float* w1  = w0 + N_EX;
        float s0 = 0.f, s1 = 0.f;
        for (int k = 0; k < N_EX; ++k) {
            float ev = exb[k];
            s0 = fmaf(ev, w0[k], s0);
            s1 = fmaf(ev, w1[k], s1);
        }
        float sv = su[b];
        xw[t] = sv * s0 + (1.0f - sv) * s1;
    } else if (idx < 22528) {
        int t = idx - 18432;            // kci[b,j]
        int b = t >> 7, j = t & 127;
        const float* exb = ex + b * N_EX;
        float s = 0.f;
        for (int k = 0; k < N_EX; ++k)
            s = fmaf(exb[k], ex_graph[k * N_KC + j], s);
        kciw[t] = s;
    } else if (idx < 30720) {
        int t = idx - 22528;            // epart[j,d]
        int j = t >> 6, d = t & 63;
        float s = fpart_b[d];
        const float* we = fpart_w + d * (D_H + D_H) + D_H; // fpart_w[d, 64+k]
        for (int k = 0; k < 64; ++k)
            s = fmaf(W_kc[k * N_KC + j], we[k], s);        // e[j,k]=W_kc[k,j]
        epartw[t] = s;
    }
}

// ---------------------------------------------------------------------------
// WMMA helper: accumulate one 16x16 f32 tile, A from LDS (row-major, lda),
// B^T from global row-major weight W (B[k,n] = W[n*ldb + k]), K = 64.
// Layouts per CDNA5 ISA 7.12.2:
//   A 16x4 : lane&15 = M row;  v.x = K(2*half), v.y = K(2*half+1)
//   B 4x16 : lane&15 = N col;  v.x = K(2*half), v.y = K(2*half+1)
// ---------------------------------------------------------------------------
__device__ __forceinline__ void wmma_k64(v8f& c,
                                         const float* __restrict__ Alds, int lda, int mBase,
                                         const float* __restrict__ Wglb, int ldb, int nBase) {
    const int lane = threadIdx.x & 31;
    const int half = lane >> 4;
    const int l    = lane & 15;
    const float* arow = Alds + (mBase + l) * lda + 2 * half;
    const float* bcol = Wglb + (nBase + l) * ldb + 2 * half;
#pragma unroll
    for (int k = 0; k < 64; k += 4) {
        v2f a; a.x = arow[k];     a.y = arow[k + 1];
        v2f b; b.x = bcol[k];     b.y = bcol[k + 1];
        c = __builtin_amdgcn_wmma_f32_16x16x4_f32(false, a, false, b,
                                                  (short)0, c, false, false);
    }
}

// ---------------------------------------------------------------------------
// Kernel 2: one workgroup per batch element b. 256 threads = 8 waves;
// wave w owns rows 16w..16w+15 of the 128-row (N_KC) dimension.
// ---------------------------------------------------------------------------
__global__ void __launch_bounds__(256)
skt_main(const float* __restrict__ hIn,
         const float* __restrict__ tgru_wih, const float* __restrict__ tgru_whh,
         const float* __restrict__ tgru_bih, const float* __restrict__ tgru_bhh,
         const float* __restrict__ fpart_w,
         const float* __restrict__ ulin_w,   const float* __restrict__ ulin_b,
         const float* __restrict__ ugru_wih, const float* __restrict__ ugru_whh,
         const float* __restrict__ ugru_bih, const float* __restrict__ ugru_bhh,
         const float* __restrict__ kcg,      const float* __restrict__ xw,
         const float* __restrict__ kciw,     const float* __restrict__ epartw,
         float* __restrict__ out) {
    __shared__ __align__(16) float sH[N_KC * D_H];   // h -> new_h
    __shared__ __align__(16) float sA[N_KC * D_H];   // delta_h -> partj
    __shared__ __align__(16) float sB[N_KC * D_H];   // dpart -> outj
    __shared__ float sGi[D3];
    __shared__ float sKci[N_KC];
    __shared__ float sKcj[N_KC];

    const int b    = blockIdx.x;
    const int tid  = threadIdx.x;
    const int lane = tid & 31;
    const int half = lane >> 4;
    const int l    = lane & 15;
    const int mBase = (tid >> 5) * 16;

    // ---- stage h[b] into LDS via CDNA5 async tensor path (ASYNCcnt) ----
    {
        const unsigned ldsH = (unsigned)(size_t)(void*)sH;           // LDS = addr[31:0]
        unsigned long long gbase =
            (unsigned long long)(const void*)(hIn + (size_t)b * N_KC * D_H);
#pragma unroll
        for (int i = 0; i < 8; ++i) {
            unsigned elem = (tid + i * 256) * 4;                      // float index
            async_ld_b128(ldsH + elem * 4, gbase + (unsigned long long)elem * 4);
        }
    }

    // ---- warm WGP$/L2 for weights of later GEMMs while async copy flies ----
    if (tid < D3) {
        __builtin_prefetch(tgru_whh + tid * D_H, 0, 1);
        __builtin_prefetch(ugru_wih + tid * D_H, 0, 1);
        __builtin_prefetch(ugru_whh + tid * D_H, 0, 1);
        if (tid < D_H) {
            __builtin_prefetch(fpart_w + tid * 2 * D_H, 0, 1);
            __builtin_prefetch(ulin_w + tid * D_H, 0, 1);
        }
    }

    if (tid < N_KC) sKci[tid] = kciw[b * N_KC + tid];
    if (tid < D3) {                     // gi = x[b] @ tgru_wih.T + bih (shared by all rows)
        float s = tgru_bih[tid];
        const float* wr = tgru_wih + tid * D_E;
        const float* xb = xw + b * D_E;
        for (int k = 0; k < D_E; ++k) s = fmaf(xb[k], wr[k], s);
        sGi[tid] = s;
    }
    wait_async0();                      // h tile resident in LDS
    __syncthreads();
    if (tid < N_KC) {                   // kcj = kci @ kc_graph
        float s = 0.f;
        for (int i = 0; i < N_KC; ++i) s = fmaf(sKci[i], kcg[i * N_KC + tid], s);
        sKcj[tid] = s;
    }
    __syncthreads();

    // ---- GEMM1: gh = sH(128x64) @ tgru_whh^T (->128x192), fused tGRU ----
    v8f acc[12];
#pragma unroll
    for (int nt = 0; nt < 12; ++nt) {
        acc[nt] = (v8f)0.0f;
        wmma_k64(acc[nt], sH, D_H, mBase, tgru_whh, D_H, nt * 16);
    }
#pragma unroll
    for (int nt = 0; nt < 4; ++nt) {
#pragma unroll
        for (int r = 0; r < 8; ++r) {
            int row = mBase + r + half * 8;
            int d   = nt * 16 + l;
            float hr = acc[nt][r]     + tgru_bhh[d];
            float hz = acc[nt + 4][r] + tgru_bhh[D_H + d];
            float hn = acc[nt + 8][r] + tgru_bhh[2 * D_H + d];
            float rr = sigmoidf_(sGi[d] + hr);
            float zz = sigmoidf_(sGi[D_H + d] + hz);
            float nn = tanhf(sGi[2 * D_H + d] + rr * hn);
            float hv = sH[row * D_H + d];
            float th = (1.0f - zz) * nn + zz * hv;
            float ki = sKci[row];
            float nh = (1.0f - ki) * hv + ki * th;
            sH[row * D_H + d] = nh;          // new_h (own rows only)
            sA[row * D_H + d] = nh - hv;     // delta_h
        }
    }
    // ---- GEMM2: dpart = delta @ W_dh^T  (W_dh[n,k] = fpart_w[n*128+k]) ----
#pragma unroll
    for (int nt = 0; nt < 4; ++nt) {
        v8f c = (v8f)0.0f;
        wmma_k64(c, sA, D_H, mBase, fpart_w, 2 * D_H, nt * 16);
#pragma unroll
        for (int r = 0; r < 8; ++r)
            sB[(mBase + r + half * 8) * D_H + nt * 16 + l] = c[r];
    }
    __syncthreads();

    // ---- partj[j,:] = sum_i kcg[i,j]*kci[i]*relu(dpart[i,:]+epart[j,:]) ----
    {
        const int j     = tid >> 1;
        const int dbase = (tid & 1) * 32;
        float ep[32], accp[32];
        const float* eprow = epartw + j * D_H + dbase;
#pragma unroll
        for (int dd = 0; dd < 32; ++dd) { ep[dd] = eprow[dd]; accp[dd] = 0.f; }
        for (int i = 0; i < N_KC; ++i) {
            float w = kcg[i * N_KC + j] * sKci[i];
            const float* dp = sB + i * D_H + dbase;
#pragma unroll
            for (int dd = 0; dd < 32; ++dd) {
                float p = dp[dd] + ep[dd];
                accp[dd] = fmaf(w, fmaxf(p, 0.f), accp[dd]);
            }
        }
#pragma unroll
        for (int dd = 0; dd < 32; ++dd) sA[j * D_H + dbase + dd] = accp[dd];
    }
    __syncthreads();

    // ---- GEMM3: outj = relu(partj @ ulin_w^T + ulin_b) ----
#pragma unroll
    for (int nt = 0; nt < 4; ++nt) {
        v8f c = (v8f)0.0f;
        wmma_k64(c, sA, D_H, mBase, ulin_w, D_H, nt * 16);
#pragma unroll
        for (int r = 0; r < 8; ++r) {
            int d = nt * 16 + l;
            sB[(mBase + r + half * 8) * D_H + d] = fmaxf(c[r] + ulin_b[d], 0.f);
        }
    }
    __syncthreads();

    // ---- GEMM4/5: gi2 = outj @ ugru_wih^T, gh2 = new_h @ ugru_whh^T, fused uGRU ----
    v8f acc2[12];
#pragma unroll
    for (int nt = 0; nt < 12; ++nt) {
        acc[nt] = (v8f)0.0f;
        wmma_k64(acc[nt], sB, D_H, mBase, ugru_wih, D_H, nt * 16);
    }
#pragma unroll
    for (int nt = 0; nt < 12; ++nt) {
        acc2[nt] = (v8f)0.0f;
        wmma_k64(acc2[nt], sH, D_H, mBase, ugru_whh, D_H, nt * 16);
    }
#pragma unroll
    for (int nt = 0; nt < 4; ++nt) {
#pragma unroll
        for (int r = 0; r < 8; ++r) {
            int row = mBase + r + half * 8;
            int d   = nt * 16 + l;
            float ir = acc[nt][r]      + ugru_bih[d];
            float iz = acc[nt + 4][r]  + ugru_bih[D_H + d];
            float in = acc[nt + 8][r]  + ugru_bih[2 * D_H + d];
            float hr = acc2[nt][r]     + ugru_bhh[d];
            float hz = acc2[nt + 4][r] + ugru_bhh[D_H + d];
            float hn = acc2[nt + 8][r] + ugru_bhh[2 * D_H + d];
            float rr = sigmoidf_(ir + hr);
            float zz = sigmoidf_(iz + hz);
            float nn = tanhf(in + rr * hn);
            float nh = sH[row * D_H + d];
            float uh = (1.0f - zz) * nn + zz * nh;
            float kj = sKcj[row];
            out[(size_t)b * N_KC * D_H + row * D_H + d] = (1.0f - kj) * nh + kj * uh;
        }
    }
}

extern "C" void kernel_launch(void* const* d_in, const int* in_sizes, int n_in,
                              void* d_out, int out_size, void* d_ws, size_t ws_size,
                              hipStream_t stream) {
    const float* h        = (const float*)d_in[0];
    const float* ex       = (const float*)d_in[1];
    const float* su       = (const float*)d_in[2];
    const float* ex_graph = (const float*)d_in[3];
    const float* kc_gamma = (const float*)d_in[4];
    const float* W_ex     = (const float*)d_in[5];
    const float* W_kc     = (const float*)d_in[6];
    const float* tgru_wih = (const float*)d_in[7];
    const float* tgru_whh = (const float*)d_in[8];
    const float* tgru_bih = (const float*)d_in[9];
    const float* tgru_bhh = (const float*)d_in[10];
    const float* fpart_w  = (const float*)d_in[11];
    const float* fpart_b  = (const float*)d_in[12];
    const float* ulin_w   = (const float*)d_in[13];
    const float* ulin_b   = (const float*)d_in[14];
    const float* ugru_wih = (const float*)d_in[15];
    const float* ugru_whh = (const float*)d_in[16];
    const float* ugru_bih = (const float*)d_in[17];
    const float* ugru_bhh = (const float*)d_in[18];

    float* ws     = (float*)d_ws;
    float* kcg    = ws;            // 16384
    float* xw     = ws + 16384;    // 2048
    float* kciw   = ws + 18432;    // 4096
    float* epartw = ws + 22528;    // 8192

    skt_pre<<<120, 256, 0, stream>>>(kc_gamma, ex, su, W_ex, ex_graph, W_kc,
                                     fpart_w, fpart_b, kcg, xw, kciw, epartw);
    skt_main<<<BATCH, 256, 0, stream>>>(h, tgru_wih, tgru_whh, tgru_bih, tgru_bhh,
                                        fpart_w, ulin_w, ulin_b,
                                        ugru_wih, ugru_whh, ugru_bih, ugru_bhh,
                                        kcg, xw, kciw, epartw, (float*)d_out);
}